// GIN_Net_72353019068534
// MI455X (gfx1250) — compile-verified
//
#include <hip/hip_runtime.h>
#include <hip/hip_bf16.h>

// ---------------------------------------------------------------------------
// GIN 2-layer GNN inference for MI455X (gfx1250, wave32, WMMA).
//   agg[i] = sum_{(j->i)} x[j]            (L2-resident f32 atomics)
//   h      = relu( relu((x+agg)@Wa+ba) @ Wb + bb )   fused, f32 WMMA 16x16x4
//   out    = log_softmax( h2 @ Wfc + bfc )
// ---------------------------------------------------------------------------

typedef __attribute__((ext_vector_type(2))) float v2f;
typedef __attribute__((ext_vector_type(8))) float v8f;

#define N_NODES 100000
#define N_EDGES 1600000
#define HIDDEN  128
#define NCLS    40
#define BM      64           // rows per block in the fused MLP kernel

// LDS strides (in floats) chosen for bank-conflict-free WMMA fragment loads.
#define SIN_LD  132          // input/t tile stride  (A frags: ds_load_b64, 4m pattern)
#define SW_LD   136          // weight tile stride   (B frags: half-wave offset = 16 banks)

// ---------------------------------------------------------------------------
// Edge aggregation: one wave per edge, lane l moves float4 l of the 128-float
// feature row with global f32 atomics (agg is L2-resident: 51.2 MB < 192 MB).
// ---------------------------------------------------------------------------
__global__ void __launch_bounds__(256)
scatter_add_kernel(const float* __restrict__ feat,
                   const long long* __restrict__ edge_index,  // [2, N_EDGES] int64
                   float* __restrict__ agg, int n_edges) {
    const int lane = threadIdx.x & 31;
    const long long e = (long long)blockIdx.x * 8 + (threadIdx.x >> 5);
    if (e >= n_edges) return;
    const long long s = edge_index[e];
    const long long d = edge_index[(long long)n_edges + e];
    const float4 v = ((const float4*)(feat + s * HIDDEN))[lane];
    float* base = agg + d * HIDDEN + lane * 4;
    atomicAdd(base + 0, v.x);
    atomicAdd(base + 1, v.y);
    atomicAdd(base + 2, v.z);
    atomicAdd(base + 3, v.w);
}

// ---------------------------------------------------------------------------
// Fused GIN MLP: out = relu( relu((x+agg)@Wa + ba) @ Wb + bb )
// Block = 8 waves, 64 rows. Wave j owns output columns [16j,16j+16).
// f32 WMMA 16x16x4; K = 128 => 32 WMMA per 16x16 tile per GEMM.
// ---------------------------------------------------------------------------
__device__ __forceinline__ v8f zero8() { v8f z = {}; return z; }

__global__ void __launch_bounds__(256)
gin_mlp_kernel(const float* __restrict__ xin, const float* __restrict__ agg,
               const float* __restrict__ wa, const float* __restrict__ ba,
               const float* __restrict__ wb, const float* __restrict__ bb,
               float* __restrict__ out, int M) {
    __shared__ float s_in[BM * SIN_LD];         // (x+agg) tile; reused for t
    __shared__ float s_wa[HIDDEN * SW_LD];
    __shared__ float s_wb[HIDDEN * SW_LD];
    __shared__ float s_ba[HIDDEN];
    __shared__ float s_bb[HIDDEN];

    const int tid   = threadIdx.x;
    const int lane  = tid & 31;
    const int wid   = tid >> 5;                 // 0..7 : column tile
    const int row0  = blockIdx.x * BM;

    // --- stage weights + biases (float4 global -> padded LDS) ---
    {
        const float4* wa4 = (const float4*)wa;
        const float4* wb4 = (const float4*)wb;
        for (int i = tid; i < HIDDEN * (HIDDEN / 4); i += 256) {
            const int k  = i >> 5;              // row (K)
            const int n4 = (i & 31) << 2;       // col group
            *(float4*)&s_wa[k * SW_LD + n4] = wa4[i];
            *(float4*)&s_wb[k * SW_LD + n4] = wb4[i];
        }
        if (tid < HIDDEN) { s_ba[tid] = ba[tid]; s_bb[tid] = bb[tid]; }
    }
    // --- stage input tile: (x + agg), zero-padded past M ---
    {
        const float4* x4 = (const float4*)xin;
        const float4* a4 = (const float4*)agg;
        for (int i = tid; i < BM * (HIDDEN / 4); i += 256) {
            const int r  = i >> 5;
            const int c4 = (i & 31) << 2;
            float4 v = make_float4(0.f, 0.f, 0.f, 0.f);
            const int gr = row0 + r;
            if (gr < M) {
                const size_t gi = (size_t)gr * (HIDDEN / 4) + (i & 31);
                const float4 vx = x4[gi];
                const float4 va = a4[gi];
                v = make_float4(vx.x + va.x, vx.y + va.y, vx.z + va.z, vx.w + va.w);
            }
            *(float4*)&s_in[r * SIN_LD + c4] = v;
        }
    }
    __syncthreads();

    const int m15   = lane & 15;                // A-frag / C-frag column-in-tile
    const int khalf = (lane >> 4) << 1;         // 0 | 2 (K sub-step per half-wave)
    const int n     = (wid << 4) + m15;         // output column 0..127
    const int mbase = (lane >> 4) << 3;         // C/D rows 0..7 | 8..15

    // ---------------- GEMM1: t = relu(in @ Wa + ba) ----------------
    v8f acc0 = zero8(), acc1 = zero8(), acc2 = zero8(), acc3 = zero8();
#pragma unroll
    for (int k0 = 0; k0 < HIDDEN; k0 += 4) {
        const int kb = k0 + khalf;
        v2f bf;
        bf.x = s_wa[kb * SW_LD + n];
        bf.y = s_wa[(kb + 1) * SW_LD + n];
        const v2f a0 = *(const v2f*)&s_in[(0 * 16 + m15) * SIN_LD + kb];
        const v2f a1 = *(const v2f*)&s_in[(1 * 16 + m15) * SIN_LD + kb];
        const v2f a2 = *(const v2f*)&s_in[(2 * 16 + m15) * SIN_LD + kb];
        const v2f a3 = *(const v2f*)&s_in[(3 * 16 + m15) * SIN_LD + kb];
        acc0 = __builtin_amdgcn_wmma_f32_16x16x4_f32(false, a0, false, bf, (short)0, acc0, false, false);
        acc1 = __builtin_amdgcn_wmma_f32_16x16x4_f32(false, a1, false, bf, (short)0, acc1, false, false);
        acc2 = __builtin_amdgcn_wmma_f32_16x16x4_f32(false, a2, false, bf, (short)0, acc2, false, false);
        acc3 = __builtin_amdgcn_wmma_f32_16x16x4_f32(false, a3, false, bf, (short)0, acc3, false, false);
    }
    __syncthreads();                            // all reads of s_in complete

    // bias + relu, write t back into s_in (in place)
    {
        const float bav = s_ba[n];
#pragma unroll
        for (int i = 0; i < 8; ++i) {
            s_in[(0 * 16 + mbase + i) * SIN_LD + n] = fmaxf(acc0[i] + bav, 0.f);
            s_in[(1 * 16 + mbase + i) * SIN_LD + n] = fmaxf(acc1[i] + bav, 0.f);
            s_in[(2 * 16 + mbase + i) * SIN_LD + n] = fmaxf(acc2[i] + bav, 0.f);
            s_in[(3 * 16 + mbase + i) * SIN_LD + n] = fmaxf(acc3[i] + bav, 0.f);
        }
    }
    __syncthreads();

    // ---------------- GEMM2: out = relu(t @ Wb + bb) ----------------
    acc0 = zero8(); acc1 = zero8(); acc2 = zero8(); acc3 = zero8();
#pragma unroll
    for (int k0 = 0; k0 < HIDDEN; k0 += 4) {
        const int kb = k0 + khalf;
        v2f bf;
        bf.x = s_wb[kb * SW_LD + n];
        bf.y = s_wb[(kb + 1) * SW_LD + n];
        const v2f a0 = *(const v2f*)&s_in[(0 * 16 + m15) * SIN_LD + kb];
        const v2f a1 = *(const v2f*)&s_in[(1 * 16 + m15) * SIN_LD + kb];
        const v2f a2 = *(const v2f*)&s_in[(2 * 16 + m15) * SIN_LD + kb];
        const v2f a3 = *(const v2f*)&s_in[(3 * 16 + m15) * SIN_LD + kb];
        acc0 = __builtin_amdgcn_wmma_f32_16x16x4_f32(false, a0, false, bf, (short)0, acc0, false, false);
        acc1 = __builtin_amdgcn_wmma_f32_16x16x4_f32(false, a1, false, bf, (short)0, acc1, false, false);
        acc2 = __builtin_amdgcn_wmma_f32_16x16x4_f32(false, a2, false, bf, (short)0, acc2, false, false);
        acc3 = __builtin_amdgcn_wmma_f32_16x16x4_f32(false, a3, false, bf, (short)0, acc3, false, false);
    }

    // bias + outer relu + store
    {
        const float bbv = s_bb[n];
#pragma unroll
        for (int i = 0; i < 8; ++i) {
            int gr;
            gr = row0 + 0 * 16 + mbase + i;
            if (gr < M) out[(size_t)gr * HIDDEN + n] = fmaxf(acc0[i] + bbv, 0.f);
            gr = row0 + 1 * 16 + mbase + i;
            if (gr < M) out[(size_t)gr * HIDDEN + n] = fmaxf(acc1[i] + bbv, 0.f);
            gr = row0 + 2 * 16 + mbase + i;
            if (gr < M) out[(size_t)gr * HIDDEN + n] = fmaxf(acc2[i] + bbv, 0.f);
            gr = row0 + 3 * 16 + mbase + i;
            if (gr < M) out[(size_t)gr * HIDDEN + n] = fmaxf(acc3[i] + bbv, 0.f);
        }
    }
}

// ---------------------------------------------------------------------------
// Final FC (128 -> 40) + exact log_softmax. One thread per row; weights
// broadcast from LDS (all lanes read the same word -> conflict-free).
// ---------------------------------------------------------------------------
__global__ void __launch_bounds__(256)
fc_logsoftmax_kernel(const float* __restrict__ h, const float* __restrict__ wfc,
                     const float* __restrict__ bfc, float* __restrict__ out, int M) {
    __shared__ float s_w[HIDDEN * NCLS];
    __shared__ float s_b[NCLS];
    for (int i = threadIdx.x; i < HIDDEN * NCLS; i += 256) s_w[i] = wfc[i];
    if (threadIdx.x < NCLS) s_b[threadIdx.x] = bfc[threadIdx.x];
    __syncthreads();

    const int row = blockIdx.x * 256 + threadIdx.x;
    if (row >= M) return;

    float acc[NCLS];
#pragma unroll
    for (int c = 0; c < NCLS; ++c) acc[c] = s_b[c];

    const float* hr = h + (size_t)row * HIDDEN;
    for (int k = 0; k < HIDDEN; ++k) {
        const float v = hr[k];
#pragma unroll
        for (int c = 0; c < NCLS; ++c) acc[c] = fmaf(v, s_w[k * NCLS + c], acc[c]);
    }

    float m = acc[0];
#pragma unroll
    for (int c = 1; c < NCLS; ++c) m = fmaxf(m, acc[c]);
    float s = 0.f;
#pragma unroll
    for (int c = 0; c < NCLS; ++c) s += expf(acc[c] - m);
    const float lse = m + logf(s);

    float* o = out + (size_t)row * NCLS;
#pragma unroll
    for (int c = 0; c < NCLS; ++c) o[c] = acc[c] - lse;
}

// ---------------------------------------------------------------------------
extern "C" void kernel_launch(void* const* d_in, const int* in_sizes, int n_in,
                              void* d_out, int out_size, void* d_ws, size_t ws_size,
                              hipStream_t stream) {
    (void)in_sizes; (void)n_in; (void)out_size; (void)ws_size;

    const float*     x   = (const float*)d_in[0];
    const long long* ei  = (const long long*)d_in[1];   // int64 edge_index [2, E]
    const float*     w1a = (const float*)d_in[2];
    const float*     b1a = (const float*)d_in[3];
    const float*     w1b = (const float*)d_in[4];
    const float*     b1b = (const float*)d_in[5];
    const float*     w2a = (const float*)d_in[6];
    const float*     b2a = (const float*)d_in[7];
    const float*     w2b = (const float*)d_in[8];
    const float*     b2b = (const float*)d_in[9];
    const float*     wfc = (const float*)d_in[10];
    const float*     bfc = (const float*)d_in[11];
    float*           out = (float*)d_out;

    const int M = N_NODES, E = N_EDGES;
    const size_t featN = (size_t)M * HIDDEN;

    float* agg = (float*)d_ws;          // 51.2 MB
    float* h1  = agg + featN;           // 51.2 MB
    float* h2  = h1 + featN;            // 51.2 MB

    const dim3 blk(256);
    const dim3 gEdge((E + 7) / 8);
    const dim3 gMlp((M + BM - 1) / BM);
    const dim3 gFc((M + 255) / 256);

    // ---- layer 1 ----
    hipMemsetAsync(agg, 0, featN * sizeof(float), stream);
    scatter_add_kernel<<<gEdge, blk, 0, stream>>>(x, ei, agg, E);
    gin_mlp_kernel<<<gMlp, blk, 0, stream>>>(x, agg, w1a, b1a, w1b, b1b, h1, M);

    // ---- layer 2 ----
    hipMemsetAsync(agg, 0, featN * sizeof(float), stream);
    scatter_add_kernel<<<gEdge, blk, 0, stream>>>(h1, ei, agg, E);
    gin_mlp_kernel<<<gMlp, blk, 0, stream>>>(h1, agg, w2a, b2a, w2b, b2b, h2, M);

    // ---- classifier + log_softmax ----
    fc_logsoftmax_kernel<<<gFc, blk, 0, stream>>>(h2, wfc, bfc, out, M);
}